// ExternalKnowledge_15101105013275
// MI455X (gfx1250) — compile-verified
//
#include <hip/hip_runtime.h>

// ExternalKnowledge (end-to-end memory network), MI455X / gfx1250, wave32.
// Shapes: B=32, M=512, S=10, V=32000, D=128, HOPS=3, C: [4, V, D] f32.
// d_in[0]=story(int32 [B,M,S]), d_in[1]=hidden(f32 [B,1,D]), d_in[2]=C(f32 [4,V,D])
// d_out = concat(prob_logit [B,M], u [B,D]) f32.
// d_ws: needs 4*B*M*D*4 = 32 MB for the precomputed bag-sums E[4][B][M][D].

#define HOPS 3
#define B_   32
#define M_   512
#define S_   10
#define V_   32000
#define D_   128

typedef __attribute__((ext_vector_type(2))) float v2f;
typedef __attribute__((ext_vector_type(8))) float v8f;

// ---------------------------------------------------------------- init u ----
__global__ void init_u_kernel(const float* __restrict__ hidden, float* __restrict__ u) {
    int i = blockIdx.x * blockDim.x + threadIdx.x;
    if (i < B_ * D_) u[i] = hidden[i];   // hidden is [B,1,D] contiguous
}

// ------------------------------------------------------------- bag sums -----
// E[t][b][m][d] = sum_s C[t][ story[b][m][s] ][d], for t = 0..3.
// One wave32 per output row; lane l owns float4 at d = 4*l.
// Each gathered 512B embedding row is read fully coalesced by the wave.
__global__ __launch_bounds__(256) void bag_sum_kernel(const int* __restrict__ story,
                                                      const float* __restrict__ C,
                                                      float* __restrict__ E) {
    int wave = blockIdx.x * (blockDim.x >> 5) + (threadIdx.x >> 5); // 0 .. 4*B*M-1
    int lane = threadIdx.x & 31;
    int t    = wave >> 14;        // table index (B*M = 16384 rows per table)
    int rem  = wave & 16383;      // b*M + m
    const int*   toks = story + rem * S_;
    const float* Ct   = C + (size_t)t * V_ * D_;

    float4 acc = make_float4(0.f, 0.f, 0.f, 0.f);
#pragma unroll
    for (int s = 0; s < S_; ++s) {
        int tok = toks[s];  // wave-uniform broadcast load
        const float4* row = reinterpret_cast<const float4*>(Ct + (size_t)tok * D_);
        float4 v = row[lane];
        acc.x += v.x; acc.y += v.y; acc.z += v.z; acc.w += v.w;
    }
    reinterpret_cast<float4*>(E + (size_t)wave * D_)[lane] = acc;
}

// ------------------------------------------------------------------ hop -----
// One block (512 threads = 16 waves) per batch b.
//  A) logits[m] = dot(E_h[b][m], u[b])      via V_WMMA_F32_16X16X4_F32
//  B) prob = softmax(logits) over M
//  C) o[d] = sum_m prob[m] * E_c[b][m][d];  u[b] += o
__global__ __launch_bounds__(512) void hop_kernel(const float* __restrict__ Eh,
                                                  const float* __restrict__ Ec,
                                                  float* __restrict__ u_g,
                                                  float* __restrict__ logit_out) {
    __shared__ float u_s[D_];
    __shared__ float logits[M_];     // later reused to hold softmax probs
    __shared__ float red[512];
    __shared__ float opart[4 * D_];

    const int b   = blockIdx.x;
    const int tid = threadIdx.x;
    const int lane = tid & 31;
    const int wave = tid >> 5;

    const float* Ehb = Eh + (size_t)b * M_ * D_;
    const float* Ecb = Ec + (size_t)b * M_ * D_;

    if (tid < D_) u_s[tid] = u_g[b * D_ + tid];
    __syncthreads();

    // ---- Phase A: logits via WMMA (each wave does two 16-row m-tiles) ----
#if __has_builtin(__builtin_amdgcn_wmma_f32_16x16x4_f32)
    {
        const int khalf = (lane >> 4) * 2;          // 0 for lanes 0-15, 2 for 16-31
        for (int rep = 0; rep < 2; ++rep) {
            const int m0  = (wave * 2 + rep) * 16;
            const int row = m0 + (lane & 15);
            const float* arow = Ehb + (size_t)row * D_;
            v8f c = {};                             // f32 accumulator, 8 VGPRs
            // A: 16x4 f32 tile (ISA layout: V0=K+0/K+2, V1=K+1/K+3 by lane half)
            // B: u broadcast across all 16 N columns -> every out column = matvec
#pragma unroll 4
            for (int k0 = 0; k0 < D_; k0 += 4) {
                v2f a, bb;
                a.x  = arow[k0 + khalf];
                a.y  = arow[k0 + khalf + 1];
                bb.x = u_s[k0 + khalf];
                bb.y = u_s[k0 + khalf + 1];
                c = __builtin_amdgcn_wmma_f32_16x16x4_f32(
                        false, a, false, bb, (short)0, c, false, false);
            }
            // C/D layout: VGPR j = row M=j (lanes 0-15) / M=j+8 (lanes 16-31).
            // Columns identical; lanes 0 and 16 write the 16 logits.
            if ((lane & 15) == 0) {
                const int mbase = m0 + ((lane >> 4) ? 8 : 0);
#pragma unroll
                for (int j = 0; j < 8; ++j) logits[mbase + j] = c[j];
            }
        }
    }
#else
    {   // scalar fallback: one thread per m
        float acc = 0.f;
        const float* arow = Ehb + (size_t)tid * D_;
        for (int k = 0; k < D_; ++k) acc += arow[k] * u_s[k];
        logits[tid] = acc;
    }
#endif
    __syncthreads();

    // emit logits (last hop's launch overwrites -> final prob_logit)
    logit_out[b * M_ + tid] = logits[tid];

    // ---- Phase B: softmax over M=512 ----
    float l = logits[tid];
    red[tid] = l;
    __syncthreads();
    for (int s = 256; s > 0; s >>= 1) {
        if (tid < s) red[tid] = fmaxf(red[tid], red[tid + s]);
        __syncthreads();
    }
    const float mx = red[0];
    __syncthreads();
    const float e = __expf(l - mx);
    red[tid] = e;
    __syncthreads();
    for (int s = 256; s > 0; s >>= 1) {
        if (tid < s) red[tid] += red[tid + s];
        __syncthreads();
    }
    const float inv = 1.0f / red[0];
    __syncthreads();
    logits[tid] = e * inv;          // logits now holds probs
    __syncthreads();

    // ---- Phase C: o = prob^T * E_c, u += o ----
    // thread t: d = t&127, m-group g = t>>7 (128 m's each); coalesced over d.
    {
        const int d = tid & (D_ - 1);
        const int g = tid >> 7;
        const float* base = Ecb + (size_t)g * 128 * D_;
        float acc = 0.f;
#pragma unroll 4
        for (int m = 0; m < 128; ++m)
            acc += logits[g * 128 + m] * base[(size_t)m * D_ + d];
        opart[g * D_ + d] = acc;
    }
    __syncthreads();
    if (tid < D_) {
        const float o = opart[tid] + opart[D_ + tid] + opart[2 * D_ + tid] + opart[3 * D_ + tid];
        u_g[b * D_ + tid] = u_s[tid] + o;
    }
}

// -------------------------------------------------------------- launcher ----
extern "C" void kernel_launch(void* const* d_in, const int* in_sizes, int n_in,
                              void* d_out, int out_size, void* d_ws, size_t ws_size,
                              hipStream_t stream) {
    const int*   story  = (const int*)d_in[0];
    const float* hidden = (const float*)d_in[1];
    const float* C      = (const float*)d_in[2];

    float* out        = (float*)d_out;
    float* prob_logit = out;            // [B, M]
    float* u          = out + B_ * M_;  // [B, D], updated in place across hops
    float* E          = (float*)d_ws;   // [4][B][M][D] bag sums (32 MB)

    init_u_kernel<<<(B_ * D_ + 255) / 256, 256, 0, stream>>>(hidden, u);

    // 4*B*M = 65536 rows, 8 waves (rows) per 256-thread block
    bag_sum_kernel<<<(4 * B_ * M_) / 8, 256, 0, stream>>>(story, C, E);

    for (int h = 0; h < HOPS; ++h) {
        hop_kernel<<<B_, 512, 0, stream>>>(E + (size_t)h * B_ * M_ * D_,
                                           E + (size_t)(h + 1) * B_ * M_ * D_,
                                           u, prob_logit);
    }
}